// MolecularGAT_4105988735686
// MI455X (gfx1250) — compile-verified
//
#include <hip/hip_runtime.h>

typedef __attribute__((ext_vector_type(2))) float v2f;
typedef __attribute__((ext_vector_type(8))) float v8f;
typedef __attribute__((ext_vector_type(4))) unsigned u32x4;
typedef __attribute__((ext_vector_type(4))) int i32x4;
typedef __attribute__((ext_vector_type(8))) int i32x8;

#if defined(__has_builtin)
#if __has_builtin(__builtin_amdgcn_tensor_load_to_lds)
#define HAVE_TDM 1
#endif
#if __has_builtin(__builtin_amdgcn_s_wait_tensorcnt)
#define HAVE_WAIT_TENSORCNT 1
#endif
#endif

// ---------------------------------------------------------------------------
// Utility kernels
// ---------------------------------------------------------------------------
__global__ void k_zero_f32(float* __restrict__ p, long long n) {
  long long i = (long long)blockIdx.x * blockDim.x + threadIdx.x;
  if (i < n) p[i] = 0.0f;
}

__global__ void k_init_mbits(unsigned* __restrict__ p, long long n) {
  long long i = (long long)blockIdx.x * blockDim.x + threadIdx.x;
  if (i < n) p[i] = 0x007FFFFFu;  // order-preserving encoding of -inf
}

// ---------------------------------------------------------------------------
// GEMM: C[N x Nout] = A[N x K] @ B[K x Nout], f32, via V_WMMA_F32_16X16X4_F32.
// Block = 256 threads = 8 waves; all waves share colBase = blockIdx.y*64.
// B is staged in 128-row chunks into LDS (32 KB) via the Tensor Data Mover
// (wave 0 issues tensor_load_to_lds, waits TENSORcnt, then workgroup barrier);
// a cooperative-copy fallback is used if the TDM builtin is unavailable.
// Fragment layouts (ISA 7.12.2):
//   A 16x4:  lane group g=lane>>4; element j holds (M=lane&15, K=k0+2g+j)
//   B 4x16:  element j holds (K=k0+2g+j, N=ct*16+(lane&15))  [read from LDS]
//   C 16x16: VGPR j holds (M=rowBase+j+8g, N=colBase+ct*16+(lane&15))
// ---------------------------------------------------------------------------
#define KCHUNK 128

__global__ void k_gemm_wmma(const float* __restrict__ A, const float* __restrict__ W,
                            float* __restrict__ C, int rowTiles, int K, int Nout) {
  __shared__ float Bs[KCHUNK * 64];

  const int tid  = threadIdx.x;
  const int lane = tid & 31;
  const int wave = tid >> 5;
  const int rowTile = blockIdx.x * 8 + wave;
  const int colBase = blockIdx.y * 64;
  const int g  = lane >> 4;   // 0 or 1
  const int ll = lane & 15;
  const int row = rowTile * 16 + ll;
  const bool active = (rowTile < rowTiles);   // wave-uniform

  v8f acc0 = {0,0,0,0,0,0,0,0};
  v8f acc1 = {0,0,0,0,0,0,0,0};
  v8f acc2 = {0,0,0,0,0,0,0,0};
  v8f acc3 = {0,0,0,0,0,0,0,0};

  const float* __restrict__ arow = A + (long long)row * K;

  for (int kc = 0; kc < K; kc += KCHUNK) {
    // ---- stage B[kc..kc+127][colBase..colBase+63] into LDS ----
#if defined(HAVE_TDM)
    if (wave == 0) {
      const unsigned ldsAddr = (unsigned)(unsigned long long)(const void*)Bs;
      const unsigned long long ga =
          (unsigned long long)(const void*)(W + (long long)kc * Nout + colBase);
      u32x4 g0;
      g0.x = 1u;                                    // count=1, user mode, no gather
      g0.y = ldsAddr;                               // LDS byte address
      g0.z = (unsigned)(ga & 0xFFFFFFFFu);          // global_addr[31:0]
      g0.w = (unsigned)((ga >> 32) & 0x1FFFFFFu)    // global_addr[56:32]
             | (2u << 30);                          // type = 2 ("image")
      i32x8 g1;
      g1[0] = (int)(2u << 16);                      // workgroup_mask=0, data_size=4B
      g1[1] = (int)(((unsigned)Nout & 0xFFFFu) << 16);        // tensor_dim0 lo16
      g1[2] = (int)((((unsigned)Nout >> 16) & 0xFFFFu)        // tensor_dim0 hi16
                    | (((unsigned)KCHUNK & 0xFFFFu) << 16));  // tensor_dim1 lo16
      g1[3] = (int)((64u) << 16);                   // tensor_dim1 hi16=0 | tile_dim0=64
      g1[4] = (int)(unsigned)KCHUNK;                // tile_dim1=128, tile_dim2=0
      g1[5] = (int)(unsigned)Nout;                  // tensor_dim0_stride lo32
      g1[6] = 0;                                    // stride hi16 | dim1_stride lo16
      g1[7] = 0;
      i32x4 gz4 = {0, 0, 0, 0};
#if __clang_major__ >= 23
      i32x8 gz8 = {0, 0, 0, 0, 0, 0, 0, 0};
      __builtin_amdgcn_tensor_load_to_lds(g0, g1, gz4, gz4, gz8, 0);
#else
      __builtin_amdgcn_tensor_load_to_lds(g0, g1, gz4, gz4, 0);
#endif
#if defined(HAVE_WAIT_TENSORCNT)
      __builtin_amdgcn_s_wait_tensorcnt(0);
#else
      asm volatile("s_wait_tensorcnt 0x0" ::: "memory");
#endif
    }
#else
    for (int idx = tid; idx < KCHUNK * 64; idx += 256) {
      Bs[idx] = W[(long long)(kc + (idx >> 6)) * Nout + colBase + (idx & 63)];
    }
#endif
    __syncthreads();

    if (active) {
      const float* __restrict__ ap = arow + kc;
      for (int k0 = 0; k0 < KCHUNK; k0 += 4) {
        v2f a = *(const v2f*)(ap + k0 + 2 * g);          // contiguous pair
        const float* __restrict__ r0 = &Bs[(k0 + 2 * g) * 64 + ll];
        const float* __restrict__ r1 = r0 + 64;
        v2f b;
        b.x = r0[0];  b.y = r1[0];
        acc0 = __builtin_amdgcn_wmma_f32_16x16x4_f32(false, a, false, b, (short)0, acc0, false, false);
        b.x = r0[16]; b.y = r1[16];
        acc1 = __builtin_amdgcn_wmma_f32_16x16x4_f32(false, a, false, b, (short)0, acc1, false, false);
        b.x = r0[32]; b.y = r1[32];
        acc2 = __builtin_amdgcn_wmma_f32_16x16x4_f32(false, a, false, b, (short)0, acc2, false, false);
        b.x = r0[48]; b.y = r1[48];
        acc3 = __builtin_amdgcn_wmma_f32_16x16x4_f32(false, a, false, b, (short)0, acc3, false, false);
      }
    }
    __syncthreads();
  }

  if (active) {
#pragma unroll
    for (int j = 0; j < 8; j++) {
      const int m = rowTile * 16 + j + 8 * g;
      float* __restrict__ cp = C + (long long)m * Nout + colBase + ll;
      cp[0]  = acc0[j];
      cp[16] = acc1[j];
      cp[32] = acc2[j];
      cp[48] = acc3[j];
    }
  }
}

// ---------------------------------------------------------------------------
// Attention logits: al_src[n,h] = sum_c h[n, h*64+c] * a_src[h,c] (same for dst)
// One wave per node; per-head partial sums reduced with wave32 shuffles.
// ---------------------------------------------------------------------------
__global__ void k_attn_logits(const float* __restrict__ h,
                              const float* __restrict__ a_src,
                              const float* __restrict__ a_dst,
                              float* __restrict__ alS, float* __restrict__ alD,
                              int N, int heads) {
  const int wave = (blockIdx.x * blockDim.x + threadIdx.x) >> 5;
  const int lane = threadIdx.x & 31;
  if (wave >= N) return;
  const int HC = heads * 64;
  float ss[4] = {0.f, 0.f, 0.f, 0.f};
  float sd[4] = {0.f, 0.f, 0.f, 0.f};
  const float* __restrict__ hr = h + (long long)wave * HC;
  for (int i = lane; i < HC; i += 32) {
    const int hh = i >> 6;
    const float v = hr[i];
    ss[hh] += v * a_src[i];
    sd[hh] += v * a_dst[i];
  }
  for (int hh = 0; hh < heads; hh++) {
    float a = ss[hh], b = sd[hh];
    for (int off = 16; off > 0; off >>= 1) {
      a += __shfl_down(a, off, 32);
      b += __shfl_down(b, off, 32);
    }
    if (lane == 0) {
      alS[(long long)wave * heads + hh] = a;
      alD[(long long)wave * heads + hh] = b;
    }
  }
}

// ---------------------------------------------------------------------------
// Segment softmax over incoming edges (E real edges + N self-loops)
// ---------------------------------------------------------------------------
__device__ __forceinline__ unsigned enc_ord(float f) {
  unsigned u = __float_as_uint(f);
  return (u & 0x80000000u) ? ~u : (u | 0x80000000u);
}
__device__ __forceinline__ float dec_ord(unsigned e) {
  unsigned u = (e & 0x80000000u) ? (e ^ 0x80000000u) : ~e;
  return __uint_as_float(u);
}

__global__ void k_edge_max(const int* __restrict__ ei, int E, int N,
                           const float* __restrict__ alS, const float* __restrict__ alD,
                           unsigned* __restrict__ mbits, int heads) {
  const int i = blockIdx.x * blockDim.x + threadIdx.x;
  if (i >= E + N) return;
  const int s = (i < E) ? ei[i]     : (i - E);
  const int d = (i < E) ? ei[E + i] : (i - E);
  for (int hh = 0; hh < heads; hh++) {
    float e = alS[(long long)s * heads + hh] + alD[(long long)d * heads + hh];
    e = (e > 0.0f) ? e : 0.2f * e;
    atomicMax(&mbits[(long long)d * heads + hh], enc_ord(e));
  }
}

__global__ void k_edge_expsum(const int* __restrict__ ei, int E, int N,
                              const float* __restrict__ alS, const float* __restrict__ alD,
                              const unsigned* __restrict__ mbits,
                              float* __restrict__ denom, float* __restrict__ exbuf,
                              int heads) {
  const int i = blockIdx.x * blockDim.x + threadIdx.x;
  if (i >= E + N) return;
  const int s = (i < E) ? ei[i]     : (i - E);
  const int d = (i < E) ? ei[E + i] : (i - E);
  for (int hh = 0; hh < heads; hh++) {
    float e = alS[(long long)s * heads + hh] + alD[(long long)d * heads + hh];
    e = (e > 0.0f) ? e : 0.2f * e;
    const float m = dec_ord(mbits[(long long)d * heads + hh]);
    const float ex = __expf(e - m);
    exbuf[(long long)i * heads + hh] = ex;
    atomicAdd(&denom[(long long)d * heads + hh], ex);
  }
}

__global__ void k_edge_agg(const int* __restrict__ ei, int E, int N,
                           const float* __restrict__ hsrc,
                           const float* __restrict__ exbuf,
                           const float* __restrict__ denom,
                           float* __restrict__ agg, int heads) {
  const long long widx = ((long long)blockIdx.x * blockDim.x + threadIdx.x) >> 5;
  const int lane = threadIdx.x & 31;
  if (widx >= (long long)(E + N)) return;
  const int i = (int)widx;
  const int s = (i < E) ? ei[i]     : (i - E);
  const int d = (i < E) ? ei[E + i] : (i - E);
  const int HC = heads * 64;
  for (int c = lane; c < HC; c += 32) {
    const int hh = c >> 6;
    const float w = exbuf[(long long)i * heads + hh] / denom[(long long)d * heads + hh];
    atomicAdd(&agg[(long long)d * HC + c], hsrc[(long long)s * HC + c] * w);
  }
}

// ---------------------------------------------------------------------------
// Finalize: act = (elu?) (agg + bias)
// ---------------------------------------------------------------------------
__global__ void k_finalize(const float* __restrict__ agg, const float* __restrict__ bias,
                           float* __restrict__ out, long long total, int HC, int do_elu) {
  const long long i = (long long)blockIdx.x * blockDim.x + threadIdx.x;
  if (i >= total) return;
  float v = agg[i] + bias[i % HC];
  if (do_elu) v = (v > 0.0f) ? v : (__expf(v) - 1.0f);
  out[i] = v;
}

// ---------------------------------------------------------------------------
// Global mean pool accumulate + final linear (C=64 -> 1)
// ---------------------------------------------------------------------------
__global__ void k_pool(const float* __restrict__ h, const int* __restrict__ batch,
                       float* __restrict__ pooled, float* __restrict__ cnt, int N) {
  const long long i = (long long)blockIdx.x * blockDim.x + threadIdx.x;
  if (i >= (long long)N * 64) return;
  const int n = (int)(i >> 6);
  const int c = (int)(i & 63);
  const int g = batch[n];
  atomicAdd(&pooled[(long long)g * 64 + c], h[i]);
  if (c == 0) atomicAdd(&cnt[g], 1.0f);
}

__global__ void k_final(const float* __restrict__ pooled, const float* __restrict__ cnt,
                        const float* __restrict__ lw, const float* __restrict__ lb,
                        float* __restrict__ out, int G) {
  const int wave = (blockIdx.x * blockDim.x + threadIdx.x) >> 5;
  const int lane = threadIdx.x & 31;
  if (wave >= G) return;
  float s = pooled[(long long)wave * 64 + lane] * lw[lane] +
            pooled[(long long)wave * 64 + lane + 32] * lw[lane + 32];
  for (int off = 16; off > 0; off >>= 1) s += __shfl_down(s, off, 32);
  if (lane == 0) {
    float c = cnt[wave];
    c = (c > 1.0f) ? c : 1.0f;
    out[wave] = s / c + lb[0];
  }
}

// ---------------------------------------------------------------------------
// Host launch
// ---------------------------------------------------------------------------
static inline int blks(long long n, int t) { return (int)((n + t - 1) / t); }

extern "C" void kernel_launch(void* const* d_in, const int* in_sizes, int n_in,
                              void* d_out, int out_size, void* d_ws, size_t ws_size,
                              hipStream_t stream) {
  const float* x     = (const float*)d_in[0];
  const int*   ei    = (const int*)d_in[1];
  const int*   batch = (const int*)d_in[2];
  const float* W1  = (const float*)d_in[3];
  const float* a1s = (const float*)d_in[4];
  const float* a1d = (const float*)d_in[5];
  const float* b1  = (const float*)d_in[6];
  const float* W2  = (const float*)d_in[7];
  const float* a2s = (const float*)d_in[8];
  const float* a2d = (const float*)d_in[9];
  const float* b2  = (const float*)d_in[10];
  const float* W3  = (const float*)d_in[11];
  const float* a3s = (const float*)d_in[12];
  const float* a3d = (const float*)d_in[13];
  const float* b3  = (const float*)d_in[14];
  const float* lw  = (const float*)d_in[15];
  const float* lb  = (const float*)d_in[16];

  const int N = in_sizes[0] / 128;
  const int E = in_sizes[1] / 2;
  const int G = out_size;
  const int T = E + N;

  // Workspace carve-up
  char* ws = (char*)d_ws;
  float* bufH   = (float*)ws; ws += (size_t)N * 256 * sizeof(float);
  float* bufA   = (float*)ws; ws += (size_t)N * 256 * sizeof(float);
  float* bufAgg = (float*)ws; ws += (size_t)N * 256 * sizeof(float);
  float* alS    = (float*)ws; ws += (size_t)N * 4 * sizeof(float);
  float* alD    = (float*)ws; ws += (size_t)N * 4 * sizeof(float);
  unsigned* mbits = (unsigned*)ws; ws += (size_t)N * 4 * sizeof(unsigned);
  float* denom  = (float*)ws; ws += (size_t)N * 4 * sizeof(float);
  float* exbuf  = (float*)ws; ws += (size_t)T * 4 * sizeof(float);
  float* pooled = (float*)ws; ws += (size_t)G * 64 * sizeof(float);
  float* cnt    = (float*)ws; ws += (size_t)G * sizeof(float);

  struct LayerP { const float* W; const float* as; const float* ad; const float* b; int K; int heads; };
  LayerP L[3] = {
    { W1, a1s, a1d, b1, 128, 4 },
    { W2, a2s, a2d, b2, 256, 4 },
    { W3, a3s, a3d, b3, 256, 1 },
  };

  const float* act = x;
  for (int li = 0; li < 3; li++) {
    const int heads = L[li].heads;
    const int HC = heads * 64;
    const int Nout = HC;
    const int K = L[li].K;
    const int rowTiles = (N + 15) / 16;

    // GEMM: bufH = act @ W   (WMMA f32 16x16x4, TDM-staged B tile)
    dim3 gg((rowTiles + 7) / 8, Nout / 64);
    k_gemm_wmma<<<gg, 256, 0, stream>>>(act, L[li].W, bufH, rowTiles, K, Nout);

    // attention logits
    k_attn_logits<<<blks((long long)N * 32, 256), 256, 0, stream>>>(
        bufH, L[li].as, L[li].ad, alS, alD, N, heads);

    // softmax state init
    k_init_mbits<<<blks((long long)N * heads, 256), 256, 0, stream>>>(mbits, (long long)N * heads);
    k_zero_f32<<<blks((long long)N * heads, 256), 256, 0, stream>>>(denom, (long long)N * heads);
    k_zero_f32<<<blks((long long)N * HC, 256), 256, 0, stream>>>(bufAgg, (long long)N * HC);

    // edge passes (E real edges + N self-loops)
    k_edge_max<<<blks(T, 256), 256, 0, stream>>>(ei, E, N, alS, alD, mbits, heads);
    k_edge_expsum<<<blks(T, 256), 256, 0, stream>>>(ei, E, N, alS, alD, mbits, denom, exbuf, heads);
    k_edge_agg<<<blks((long long)T * 32, 256), 256, 0, stream>>>(ei, E, N, bufH, exbuf, denom, bufAgg, heads);

    // bias (+ ELU on layers 0,1)
    k_finalize<<<blks((long long)N * HC, 256), 256, 0, stream>>>(
        bufAgg, L[li].b, bufA, (long long)N * HC, HC, li < 2 ? 1 : 0);
    act = bufA;
  }

  // global mean pool + final linear
  k_zero_f32<<<blks((long long)G * 64, 256), 256, 0, stream>>>(pooled, (long long)G * 64);
  k_zero_f32<<<blks(G, 256), 256, 0, stream>>>(cnt, G);
  k_pool<<<blks((long long)N * 64, 256), 256, 0, stream>>>(bufA, batch, pooled, cnt, N);
  k_final<<<blks((long long)G * 32, 256), 256, 0, stream>>>(pooled, cnt, lw, lb, (float*)d_out, G);
}